// CCRGNN_74646531605030
// MI455X (gfx1250) — compile-verified
//
#include <hip/hip_runtime.h>
#include <hip/hip_bf16.h>
#include <math.h>

typedef __attribute__((ext_vector_type(2))) float v2f;
typedef __attribute__((ext_vector_type(8))) float v8f;
typedef __attribute__((ext_vector_type(4))) int v4i;

#define NPG 39
#define DEG 8
#define EPG (NPG * DEG)   /* 312 original edges per graph */
#define EPT (EPG + NPG)   /* 351 with self loops */
#define FDIM 3280

#define RES1_OFF 39
#define RES2_OFF 351
#define RES3_OFF 2847
#define OUT0_OFF 3198
#define OUT1_OFF 3199
#define OUT2_OFF 3207
#define OUT3_OFF 3271

#if defined(__has_builtin)
#if __has_builtin(__builtin_amdgcn_global_load_async_to_lds_b128) && \
    __has_builtin(__builtin_amdgcn_s_wait_asynccnt)
#define HAVE_ASYNC_LDS 1
#endif
#endif
#ifndef HAVE_ASYNC_LDS
#define HAVE_ASYNC_LDS 0
#endif

#if HAVE_ASYNC_LDS
// builtin signature (from compiler diagnostic): global ptr is a pointer to a
// 16B int vector in the global address space; LDS side is addrspace(3).
typedef __attribute__((address_space(1))) v4i gv4i;
typedef __attribute__((address_space(3))) v4i lv4i;
__device__ __forceinline__ void async_copy16(const float* gp, float* lp) {
    __builtin_amdgcn_global_load_async_to_lds_b128((gv4i*)gp, (lv4i*)lp, 0, 0);
}
#endif

// ---------------------------------------------------------------------------
// LDS float atomic max via CAS (portable; values may be negative)
// ---------------------------------------------------------------------------
__device__ __forceinline__ void atomic_max_f32_lds(float* addr, float val) {
    int* ia = (int*)addr;
    int old = __float_as_int(*addr);
    while (__int_as_float(old) < val) {
        int assumed = old;
        old = atomicCAS(ia, assumed, __float_as_int(val));
        if (old == assumed) break;
    }
}

// ---------------------------------------------------------------------------
// One workgroup per 39-node graph. Entire GATConv (linear, attention logits,
// segment softmax over dst, weighted aggregation, bias, relu) done in LDS.
// Also writes res_k slice and segment-max out_k slice of the feature matrix.
// ---------------------------------------------------------------------------
template <int FIN, int FOUT, bool FIRST>
__global__ __launch_bounds__(256) void gat_layer_kernel(
    const float* __restrict__ xin,     // [N, FIN]
    const int* __restrict__ dstIdx,    // [E] global dst indices (no self loops)
    const float* __restrict__ W,       // [FIN, FOUT]
    const float* __restrict__ a_s,     // [FOUT]
    const float* __restrict__ a_d,     // [FOUT]
    const float* __restrict__ bias,    // [FOUT]
    float* __restrict__ hout,          // [N, FOUT] or nullptr
    float* __restrict__ F,             // [B, FDIM]
    int resOff, int outOff)
{
    __shared__ float xs[NPG * FIN];
    __shared__ float Wl[FIN * FOUT];
    __shared__ float asl[FOUT];
    __shared__ float adl[FOUT];
    __shared__ float bl[FOUT];
    __shared__ float hs[NPG * FOUT];
    __shared__ float es[NPG];
    __shared__ float ed[NPG];
    __shared__ float ev[EPT];
    __shared__ short ssrc[EPT];
    __shared__ short sdst[EPT];
    __shared__ float mx[NPG];
    __shared__ float dn[NPG];
    __shared__ float outp[NPG * FOUT];

    const int g   = blockIdx.x;
    const int tid = threadIdx.x;

    // ---- stage inputs / params, init reductions -------------------------
    for (int i = tid; i < NPG * FIN; i += 256)  xs[i] = xin[(size_t)g * NPG * FIN + i];
    for (int i = tid; i < FIN * FOUT; i += 256) Wl[i] = W[i];
    for (int i = tid; i < FOUT; i += 256) {
        asl[i] = a_s[i]; adl[i] = a_d[i]; bl[i] = bias[i];
    }
    for (int i = tid; i < NPG; i += 256) { mx[i] = -1e30f; dn[i] = 0.0f; }
    for (int i = tid; i < NPG * FOUT; i += 256) outp[i] = 0.0f;
    __syncthreads();

    // ---- h = x @ W ------------------------------------------------------
    for (int idx = tid; idx < NPG * FOUT; idx += 256) {
        const int i = idx / FOUT, f = idx % FOUT;
        float s = 0.0f;
#pragma unroll
        for (int k = 0; k < FIN; ++k) s += xs[i * FIN + k] * Wl[k * FOUT + f];
        hs[idx] = s;
    }
    __syncthreads();

    // ---- attention logit halves per node --------------------------------
    for (int i = tid; i < NPG; i += 256) {
        float s1 = 0.0f, s2 = 0.0f;
#pragma unroll
        for (int f = 0; f < FOUT; ++f) {
            s1 += hs[i * FOUT + f] * asl[f];
            s2 += hs[i * FOUT + f] * adl[f];
        }
        es[i] = s1; ed[i] = s2;
    }
    __syncthreads();

    // ---- per-edge leaky-relu score + segment max over dst ---------------
    for (int t = tid; t < EPT; t += 256) {
        int srcL, dstL;
        if (t < EPG) {
            srcL = t / DEG;
            dstL = dstIdx[(size_t)g * EPG + t] - g * NPG;
        } else {
            srcL = t - EPG; dstL = srcL;   // self loop
        }
        float e = es[srcL] + ed[dstL];
        e = (e > 0.0f) ? e : 0.2f * e;     // leaky_relu, slope 0.2
        ev[t] = e;
        ssrc[t] = (short)srcL;
        sdst[t] = (short)dstL;
        atomic_max_f32_lds(&mx[dstL], e);
    }
    __syncthreads();

    // ---- w = exp(e - m), segment sum ------------------------------------
    for (int t = tid; t < EPT; t += 256) {
        float w = expf(ev[t] - mx[sdst[t]]);
        ev[t] = w;
        atomicAdd(&dn[sdst[t]], w);
    }
    __syncthreads();

    // ---- alpha = w / (denom + eps) --------------------------------------
    for (int t = tid; t < EPT; t += 256)
        ev[t] = ev[t] / (dn[sdst[t]] + 1e-16f);
    __syncthreads();

    // ---- aggregate alpha * h[src] into dst ------------------------------
    for (int idx = tid; idx < EPT * FOUT; idx += 256) {
        const int t = idx / FOUT, f = idx % FOUT;
        atomicAdd(&outp[(int)sdst[t] * FOUT + f], ev[t] * hs[(int)ssrc[t] * FOUT + f]);
    }
    __syncthreads();

    // ---- + bias, relu; write h_out and res slice ------------------------
    for (int idx = tid; idx < NPG * FOUT; idx += 256) {
        const int i = idx / FOUT, f = idx % FOUT;
        float v = outp[idx] + bl[f];
        v = fmaxf(v, 0.0f);
        outp[idx] = v;
        if (hout) hout[((size_t)g * NPG + i) * FOUT + f] = v;
        F[(size_t)g * FDIM + resOff + i * FOUT + f] = v;
    }
    __syncthreads();

    // ---- per-graph segment max of this layer's output -------------------
    for (int f = tid; f < FOUT; f += 256) {
        float m = -1e30f;
#pragma unroll 1
        for (int i = 0; i < NPG; ++i) m = fmaxf(m, outp[i * FOUT + f]);
        F[(size_t)g * FDIM + outOff + f] = m;
    }

    if (FIRST) {
        // res0 = x reshaped, out0 = segment max of raw x (FIN == 1)
        for (int i = tid; i < NPG; i += 256)
            F[(size_t)g * FDIM + i] = xs[i];
        if (tid == 0) {
            float m = -1e30f;
            for (int i = 0; i < NPG; ++i) m = fmaxf(m, xs[i]);
            F[(size_t)g * FDIM + OUT0_OFF] = m;
        }
    }
}

// ---------------------------------------------------------------------------
// FP32 WMMA GEMM with async double-buffered LDS staging.
//   C = relu?(A[M,K] @ B[K,N] + bias[N])
// 256 threads = 8 wave32. Block tile 64M x 128N; wave tile 32x32 = 2x2 WMMA
// 16x16x4 tiles; KC=40 K-chunk (K % 40 == 0 required; 3280 and 5000 both ok).
// Staging uses GLOBAL_LOAD_ASYNC_TO_LDS_B128 (ASYNCcnt) when available:
// each wave issues exactly 8 async b128 per k-block (3 A + 5 B), so a counted
// s_wait_asynccnt(8) overlaps the next block's fill with this block's WMMAs.
// ---------------------------------------------------------------------------
#define ASTR 44   /* padded A stride: 176B = 16B-aligned, bank-conflict-free */

__device__ __forceinline__ void stage_tile(
    const float* __restrict__ A, const float* __restrict__ B,
    float* __restrict__ Asb, float* __restrict__ Bsb,
    int bm, int bn, int K, int N, int k0, int wv, int lane)
{
#if HAVE_ASYNC_LDS
    // A tile: 64 rows x 40 cols = 640 float4; 80 per wave (rows 8*wv..8*wv+7)
    {
        const int base = wv * 80;
#pragma unroll
        for (int i = 0; i < 3; ++i) {
            const int t = i * 32 + lane;
            if (t < 80) {
                const int idx = base + t;
                const int r = idx / 10, c4 = idx % 10;
                async_copy16(A + (size_t)(bm + r) * K + k0 + c4 * 4,
                             Asb + r * ASTR + c4 * 4);
            }
        }
    }
    // B tile: 40 rows x 128 cols = 1280 float4; 160 per wave (5 full issues)
    {
        const int base = wv * 160;
#pragma unroll
        for (int i = 0; i < 5; ++i) {
            const int idx = base + i * 32 + lane;
            const int r = idx >> 5, c4 = idx & 31;
            const int n = bn + c4 * 4;
            if (n < N) {   // N % 4 == 0, so chunks are all-valid or all-OOB
                async_copy16(B + (size_t)(k0 + r) * N + n,
                             Bsb + r * 128 + c4 * 4);
            }
        }
    }
#else
    const int tid = wv * 32 + lane;
    for (int idx = tid; idx < 64 * 40; idx += 256) {
        const int r = idx / 40, c = idx % 40;
        Asb[r * ASTR + c] = A[(size_t)(bm + r) * K + k0 + c];
    }
    for (int idx = tid; idx < 40 * 128; idx += 256) {
        const int r = idx / 128, c = idx % 128;
        const int n = bn + c;
        Bsb[idx] = (n < N) ? B[(size_t)(k0 + r) * N + n] : 0.0f;
    }
#endif
}

__global__ __launch_bounds__(256) void gemm_wmma_f32(
    const float* __restrict__ A, const float* __restrict__ B,
    const float* __restrict__ bias, float* __restrict__ C,
    int M, int N, int K, int doRelu)
{
    constexpr int KC = 40;
    __shared__ float As[2][64 * ASTR];
    __shared__ float Bs[2][KC * 128];

    const int tid  = threadIdx.x;
    const int lane = tid & 31;
    const int wv   = tid >> 5;            // wave 0..7
    const int wm   = (wv >> 2) * 32;      // 0 / 32
    const int wn   = (wv & 3) * 32;       // 0,32,64,96
    const int row  = lane & 15;
    const int hi   = lane >> 4;           // lanes 16-31 -> +2 K, +8 M

    const int bm = blockIdx.y * 64;
    const int bn = blockIdx.x * 128;

    v8f acc[2][2] = {};

    const int nk = K / KC;

    // prologue: fill buffer 0
    stage_tile(A, B, As[0], Bs[0], bm, bn, K, N, 0, wv, lane);

    for (int kb = 0; kb < nk; ++kb) {
        const int cur = kb & 1;
        if (kb + 1 < nk) {
            // prefetch next k-block into the other buffer
            stage_tile(A, B, As[cur ^ 1], Bs[cur ^ 1],
                       bm, bn, K, N, (kb + 1) * KC, wv, lane);
#if HAVE_ASYNC_LDS
            __builtin_amdgcn_s_wait_asynccnt(8);   // current block done
#endif
        } else {
#if HAVE_ASYNC_LDS
            __builtin_amdgcn_s_wait_asynccnt(0);
#endif
        }
        __syncthreads();

        const float* Asb = As[cur];
        const float* Bsb = Bs[cur];
#pragma unroll
        for (int kk = 0; kk < KC; kk += 4) {
            const int kA = kk + 2 * hi;
            v2f af[2], bf[2];
#pragma unroll
            for (int tm = 0; tm < 2; ++tm) {
                const float* ap = &Asb[(wm + 16 * tm + row) * ASTR + kA];
                v2f a; a.x = ap[0]; a.y = ap[1];
                af[tm] = a;
            }
#pragma unroll
            for (int tn = 0; tn < 2; ++tn) {
                const int col = wn + 16 * tn + row;
                v2f b; b.x = Bsb[kA * 128 + col]; b.y = Bsb[(kA + 1) * 128 + col];
                bf[tn] = b;
            }
#pragma unroll
            for (int tm = 0; tm < 2; ++tm)
#pragma unroll
                for (int tn = 0; tn < 2; ++tn)
                    acc[tm][tn] = __builtin_amdgcn_wmma_f32_16x16x4_f32(
                        false, af[tm], false, bf[tn],
                        (short)0, acc[tm][tn], false, false);
        }
        __syncthreads();
    }

    // ---- epilogue: bias + relu + store (C/D layout §7.12.2) -------------
#pragma unroll
    for (int tm = 0; tm < 2; ++tm) {
#pragma unroll
        for (int tn = 0; tn < 2; ++tn) {
            const int cc = bn + wn + 16 * tn + row;
            if (cc < N) {
                const float bv = bias[cc];
#pragma unroll
                for (int v = 0; v < 8; ++v) {
                    const int rr = bm + wm + 16 * tm + v + 8 * hi;
                    float val = acc[tm][tn][v] + bv;
                    if (doRelu) val = fmaxf(val, 0.0f);
                    C[(size_t)rr * N + cc] = val;
                }
            }
        }
    }
}

// ---------------------------------------------------------------------------
// Tiny final layer: out[4096,9] = G2[4096,1024] @ lW3[1024,9] + lb3
// ---------------------------------------------------------------------------
__global__ __launch_bounds__(256) void final_linear_kernel(
    const float* __restrict__ G2, const float* __restrict__ W3,
    const float* __restrict__ b3, float* __restrict__ out)
{
    __shared__ float rowbuf[1024];
    const int r = blockIdx.x;
    for (int i = threadIdx.x; i < 1024; i += 256)
        rowbuf[i] = G2[(size_t)r * 1024 + i];
    __syncthreads();
    if (threadIdx.x < 9) {
        float s = b3[threadIdx.x];
        for (int k = 0; k < 1024; ++k)
            s += rowbuf[k] * W3[k * 9 + threadIdx.x];
        out[(size_t)r * 9 + threadIdx.x] = s;
    }
}

// ---------------------------------------------------------------------------
extern "C" void kernel_launch(void* const* d_in, const int* in_sizes, int n_in,
                              void* d_out, int out_size, void* d_ws, size_t ws_size,
                              hipStream_t stream)
{
    const float* x   = (const float*)d_in[0];
    const int*   ei  = (const int*)d_in[1];
    const float* W1  = (const float*)d_in[3];
    const float* a1s = (const float*)d_in[4];
    const float* a1d = (const float*)d_in[5];
    const float* b1  = (const float*)d_in[6];
    const float* W2  = (const float*)d_in[7];
    const float* a2s = (const float*)d_in[8];
    const float* a2d = (const float*)d_in[9];
    const float* b2  = (const float*)d_in[10];
    const float* W3  = (const float*)d_in[11];
    const float* a3s = (const float*)d_in[12];
    const float* a3d = (const float*)d_in[13];
    const float* b3  = (const float*)d_in[14];
    const float* lW1 = (const float*)d_in[15];
    const float* lb1 = (const float*)d_in[16];
    const float* lW2 = (const float*)d_in[17];
    const float* lb2 = (const float*)d_in[18];
    const float* lW3 = (const float*)d_in[19];
    const float* lb3 = (const float*)d_in[20];

    const int N = in_sizes[0];          // 159744 nodes (x is [N,1])
    const int E = in_sizes[1] / 2;      // 1277952 edges
    const int B = N / NPG;              // 4096 graphs
    const int* dst = ei + E;            // dst row of edge_index

    // workspace carve-out
    char* ws = (char*)d_ws;
    float* F  = (float*)ws;                     size_t o = (size_t)B * FDIM * 4;
    float* G1 = (float*)(ws + o);               o += (size_t)B * 5000 * 4;
    float* G2 = (float*)(ws + o);               o += (size_t)B * 1024 * 4;
    float* h1 = (float*)(ws + o);               o += (size_t)N * 8 * 4;
    float* h2 = (float*)(ws + o);

    gat_layer_kernel<1, 8, true><<<B, 256, 0, stream>>>(
        x, dst, W1, a1s, a1d, b1, h1, F, RES1_OFF, OUT1_OFF);
    gat_layer_kernel<8, 64, false><<<B, 256, 0, stream>>>(
        h1, dst, W2, a2s, a2d, b2, h2, F, RES2_OFF, OUT2_OFF);
    gat_layer_kernel<64, 9, false><<<B, 256, 0, stream>>>(
        h2, dst, W3, a3s, a3d, b3, nullptr, F, RES3_OFF, OUT3_OFF);

    // f @ lW1 (+relu): [4096,3280] x [3280,5000]
    gemm_wmma_f32<<<dim3((5000 + 127) / 128, B / 64), 256, 0, stream>>>(
        F, lW1, lb1, G1, B, 5000, FDIM, 1);
    // @ lW2 (+relu): [4096,5000] x [5000,1024]
    gemm_wmma_f32<<<dim3((1024 + 127) / 128, B / 64), 256, 0, stream>>>(
        G1, lW2, lb2, G2, B, 1024, 5000, 1);
    // @ lW3: [4096,1024] x [1024,9]
    final_linear_kernel<<<B, 256, 0, stream>>>(G2, lW3, lb3, (float*)d_out);
}